// GraphSAGELayer_65635690218034
// MI455X (gfx1250) — compile-verified
//
#include <hip/hip_runtime.h>
#include <hip/hip_bf16.h>

// GraphSAGE layer for MI455X (gfx1250, wave32).
//  Phase 1: scatter-mean aggregation (memory/atomic bound, ~1.6 GB -> ~70us @ 23.3 TB/s)
//  Phase 2: fused concat + GEMM(100000x256 @ 256x256) + bias + ReLU using
//           V_WMMA_F32_16X16X4_F32 (fp32 WMMA: exact-match numerics vs fp32 reference;
//           the layer is scatter-bound so low-precision WMMA buys nothing end-to-end).
//  v2: 32-row blocks (B-fragment reuse across 2 row-tiles halves W/L2 traffic),
//      uniform fast-path epilogue (kills per-row EXEC-mask churn),
//      scalar (SMEM) edge-index loads, float4 scratch zeroing.

typedef __attribute__((ext_vector_type(2))) float v2f;
typedef __attribute__((ext_vector_type(8))) float v8f;

#define SAGE_IN_DIM 128
#define SAGE_K_DIM 256
#define SAGE_OUT_DIM 256
#define SAGE_ROWS 32          // rows of h per block
#define SAGE_LDS_STRIDE (SAGE_K_DIM + 4)   // pad: A-column reads spread over 16 banks

// ---------------------------------------------------------------- zero scratch
__global__ void sage_zero_kernel(float4* __restrict__ p, long long n4) {
    long long i = (long long)blockIdx.x * blockDim.x + threadIdx.x;
    if (i < n4) p[i] = make_float4(0.f, 0.f, 0.f, 0.f);
}

// ------------------------------------------------------- edge scatter (1 wave/edge)
// agg[dst] += x[src] (128 floats), deg[dst] += 1
__global__ __launch_bounds__(256) void sage_scatter_kernel(
    const float* __restrict__ x,
    const int* __restrict__ edge_src,
    const int* __restrict__ edge_dst,
    float* __restrict__ agg,
    float* __restrict__ deg,
    int E) {
    // wave id made provably scalar -> edge indices load via SMEM (s_load_b32)
    const int wid  = __builtin_amdgcn_readfirstlane((int)(threadIdx.x >> 5));
    const int edge = blockIdx.x * 8 + wid;
    const int lane = threadIdx.x & 31;
    if (edge >= E) return;

    const int s = edge_src[edge];
    const int d = edge_dst[edge];

    // coalesced 16B/lane load of the 512B source row
    const float4 xv = ((const float4*)(x + (size_t)s * SAGE_IN_DIM))[lane];
    float* base = agg + (size_t)d * SAGE_IN_DIM + lane * 4;
    atomicAdd(base + 0, xv.x);
    atomicAdd(base + 1, xv.y);
    atomicAdd(base + 2, xv.z);
    atomicAdd(base + 3, xv.w);
    if (lane == 0) atomicAdd(deg + d, 1.0f);
}

// --------------------------------------------- fused concat + GEMM + bias + ReLU
// Block = 256 threads (8 waves) handles 32 rows x 256 cols of output.
// Wave w owns column tiles {w, w+8} and both row tiles -> each B fragment feeds
// 2 WMMAs. K=256 -> 64 K-steps x 4 v_wmma_f32_16x16x4_f32 per wave.
__global__ __launch_bounds__(256) void sage_gemm_kernel(
    const float* __restrict__ x,
    const float* __restrict__ agg,
    const float* __restrict__ deg,
    const float* __restrict__ W,      // (256, 256) row-major: W[k*256 + n]
    const float* __restrict__ bias,   // (256,)
    float* __restrict__ out,          // (N, 256)
    int N) {
    __shared__ float hs[SAGE_ROWS][SAGE_LDS_STRIDE];   // ~33 KB

    const int i0  = blockIdx.x * SAGE_ROWS;
    const int tid = threadIdx.x;
    const bool full = (i0 + SAGE_ROWS) <= N;

    // ---- stage h = [x | mean-agg] for 32 rows into LDS
    for (int e = tid; e < SAGE_ROWS * SAGE_K_DIM; e += 256) {
        int r = e >> 8;         // row 0..31
        int c = e & 255;        // col 0..255
        int i = i0 + r;
        float v = 0.0f;
        if (full || i < N) {
            if (c < SAGE_IN_DIM) {
                v = x[(size_t)i * SAGE_IN_DIM + c];
            } else {
                int c2 = c - SAGE_IN_DIM;
                float dg = deg[i];
                float xv = x[(size_t)i * SAGE_IN_DIM + c2];
                v = (dg > 0.0f) ? (agg[(size_t)i * SAGE_IN_DIM + c2] / dg) : xv;
            }
        }
        hs[r][c] = v;
    }
    __syncthreads();

    const int wave  = tid >> 5;
    const int lane  = tid & 31;
    const int row   = lane & 15;       // A-matrix M (and B/C column within tile)
    const int khalf = lane >> 4;       // 0: K=0,1  1: K=2,3 (16x4 f32 A layout)

    const int col0 = wave * 16 + row;         // column tile  wave
    const int col1 = (wave + 8) * 16 + row;   // column tile  wave+8

    v8f acc00 = {};   // row-tile 0, col-tile 0
    v8f acc01 = {};   // row-tile 0, col-tile 1
    v8f acc10 = {};   // row-tile 1, col-tile 0
    v8f acc11 = {};   // row-tile 1, col-tile 1

    for (int k0 = 0; k0 < SAGE_K_DIM; k0 += 4) {
        const int kk = k0 + khalf * 2;

        v2f a0;                               // A rows 0..15
        a0.x = hs[row][kk];
        a0.y = hs[row][kk + 1];
        v2f a1;                               // A rows 16..31
        a1.x = hs[16 + row][kk];
        a1.y = hs[16 + row][kk + 1];

        v2f b0;                               // B 4x16: lanes0-15 K=0,1 ; lanes16-31 K=2,3
        b0.x = W[(size_t)kk * SAGE_OUT_DIM + col0];
        b0.y = W[(size_t)(kk + 1) * SAGE_OUT_DIM + col0];
        v2f b1;
        b1.x = W[(size_t)kk * SAGE_OUT_DIM + col1];
        b1.y = W[(size_t)(kk + 1) * SAGE_OUT_DIM + col1];

        acc00 = __builtin_amdgcn_wmma_f32_16x16x4_f32(
            false, a0, false, b0, (short)0, acc00, false, false);
        acc01 = __builtin_amdgcn_wmma_f32_16x16x4_f32(
            false, a0, false, b1, (short)0, acc01, false, false);
        acc10 = __builtin_amdgcn_wmma_f32_16x16x4_f32(
            false, a1, false, b0, (short)0, acc10, false, false);
        acc11 = __builtin_amdgcn_wmma_f32_16x16x4_f32(
            false, a1, false, b1, (short)0, acc11, false, false);
    }

    // ---- epilogue: C/D layout VGPR j -> M = j + 8*khalf (+16 for row-tile 1),
    //      N = lane&15. Fast path for full blocks: no per-row guards.
    const int rbase = khalf * 8;
    const float bb0 = bias[col0];
    const float bb1 = bias[col1];

    if (full) {
#pragma unroll
        for (int j = 0; j < 8; ++j) {
            size_t o0 = (size_t)(i0 + rbase + j) * SAGE_OUT_DIM;
            size_t o1 = (size_t)(i0 + 16 + rbase + j) * SAGE_OUT_DIM;
            float v;
            v = acc00[j] + bb0; out[o0 + col0] = v > 0.0f ? v : 0.0f;
            v = acc01[j] + bb1; out[o0 + col1] = v > 0.0f ? v : 0.0f;
            v = acc10[j] + bb0; out[o1 + col0] = v > 0.0f ? v : 0.0f;
            v = acc11[j] + bb1; out[o1 + col1] = v > 0.0f ? v : 0.0f;
        }
    } else {
#pragma unroll
        for (int j = 0; j < 8; ++j) {
            int i = i0 + rbase + j;
            if (i < N) {
                size_t o0 = (size_t)i * SAGE_OUT_DIM;
                float v;
                v = acc00[j] + bb0; out[o0 + col0] = v > 0.0f ? v : 0.0f;
                v = acc01[j] + bb1; out[o0 + col1] = v > 0.0f ? v : 0.0f;
            }
            int i2 = i + 16;
            if (i2 < N) {
                size_t o1 = (size_t)i2 * SAGE_OUT_DIM;
                float v;
                v = acc10[j] + bb0; out[o1 + col0] = v > 0.0f ? v : 0.0f;
                v = acc11[j] + bb1; out[o1 + col1] = v > 0.0f ? v : 0.0f;
            }
        }
    }
}

// ----------------------------------------------------------------------- launch
extern "C" void kernel_launch(void* const* d_in, const int* in_sizes, int n_in,
                              void* d_out, int out_size, void* d_ws, size_t ws_size,
                              hipStream_t stream) {
    const float* x        = (const float*)d_in[0];
    const int*   edge_src = (const int*)d_in[1];
    const int*   edge_dst = (const int*)d_in[2];
    const float* W        = (const float*)d_in[3];
    const float* bias     = (const float*)d_in[4];
    float*       out      = (float*)d_out;

    const int N = in_sizes[0] / SAGE_IN_DIM;
    const int E = in_sizes[1];

    float* agg = (float*)d_ws;                       // N*128 floats
    float* deg = agg + (size_t)N * SAGE_IN_DIM;      // N floats (128+1 -> odd tail ok)

    // 1) zero aggregation scratch (must happen every call: graph-replayed).
    //    N*(128+1) floats: 129 is odd, but N is even enough; round up in float4s
    //    and let the tail spill into unused ws (harness scratch is ours to use).
    long long nz  = (long long)N * (SAGE_IN_DIM + 1);
    long long nz4 = (nz + 3) / 4;
    int zb = (int)((nz4 + 255) / 256);
    sage_zero_kernel<<<zb, 256, 0, stream>>>((float4*)d_ws, nz4);

    // 2) edge scatter: one wave per edge, 8 waves per block
    int sb = (E + 7) / 8;
    sage_scatter_kernel<<<sb, 256, 0, stream>>>(x, edge_src, edge_dst, agg, deg, E);

    // 3) fused concat + fp32-WMMA GEMM + bias + ReLU (32 rows per block)
    int gb = (N + SAGE_ROWS - 1) / SAGE_ROWS;
    sage_gemm_kernel<<<gb, 256, 0, stream>>>(x, agg, deg, W, bias, out, N);
}